// LRN_51659866636963
// MI455X (gfx1250) — compile-verified
//
#include <hip/hip_runtime.h>

// LRN across channels, NCHW (64,192,56,56) fp32.
// out = x * (1 + (alpha/5) * sum_{k=-2..2} x_{c+k}^2)^(-beta)
//
// Memory-bound: ~308 MB traffic -> ~13us at 23.3 TB/s. Strategy:
//  - float4 vectorization along W (coalesced 512B/wave/channel-step)
//  - C split into 4 chunks of 48 (+/-2 halo) -> 784 blocks / 6272 waves
//  - sliding-window sum of squares kept in registers (2 FMA/elem)
//  - CDNA5 async DMA ring: global_load_async_to_lds_b128 (depth 8,
//    lookahead 6) + s_wait_asynccnt -> 6 x 512B loads in flight per wave
//    with no VGPR cost; statically scheduled (steady / peel / drain)
//  - ring reads through an explicit address_space(3) pointer so they
//    select real ds_load_b128 (round 2 showed generic->flat fallback)
//  - nontemporal stores for the write-once output

typedef __attribute__((ext_vector_type(4))) float f4;
typedef __attribute__((address_space(3))) const f4* lds_cf4p;

#define N_      64
#define C_      192
#define HW4_    784            // 56*56/4 float4 per channel plane
#define CHW4_   (C_ * HW4_)
#define CHUNK_  48
#define NCHUNK_ (C_ / CHUNK_)
#define DEPTH_  8              // LDS ring slots (16B per thread per slot)
#define PRE_    6              // async lookahead (outstanding copies)
#define LRN_K_  (0.0001f / 5.0f)   // alpha / local_size
#define BETA_   0.75f

__global__ __launch_bounds__(256) void lrn_kernel(const float* __restrict__ xf,
                                                  float* __restrict__ of) {
    const f4* __restrict__ xv = (const f4*)xf;
    f4* __restrict__ ov       = (f4*)of;

    // 8 slots * 256 threads * 16B = 32 KB (single __shared__ -> LDS offset 0)
    __shared__ f4 sh[DEPTH_ * 256];
    lds_cf4p shp = (lds_cf4p)&sh[0];   // type-level addrspace(3): forces DS ops

    const int col  = blockIdx.x * 256 + threadIdx.x;   // 0 .. 50175
    const int n    = col / HW4_;
    const int hw4  = col - n * HW4_;
    const int c0   = blockIdx.y * CHUNK_;
    const int c1   = c0 + CHUNK_;
    const long base4 = (long)n * CHW4_ + hw4;

    // channel range staged through LDS (always in-bounds loads)
    const int lo = (c0 - 2 < 0) ? 0 : (c0 - 2);
    const int hi = (c1 + 2 > C_) ? C_ : (c1 + 2);
    const int cs = hi - PRE_ - 1;   // drain phase starts at cs (peel at cs-1)

    int nextIssue = lo;

    // one async 16B-per-lane copy: global -> this lane's private LDS slot
#define ISSUE() do {                                                               \
        unsigned loff_ = ((unsigned)(((nextIssue - lo) & (DEPTH_ - 1)) * 256)      \
                          + threadIdx.x) * 16u;                                    \
        unsigned long long ga_ =                                                   \
            (unsigned long long)(const void*)(xv + base4 + (long)nextIssue * HW4_);\
        asm volatile("global_load_async_to_lds_b128 %0, %1, off"                   \
                     :: "v"(loff_), "v"(ga_) : "memory");                          \
        ++nextIssue;                                                               \
    } while (0)
#define WAIT_STEADY() asm volatile("s_wait_asynccnt 0x5" ::: "memory")  /* PRE_-1 */
#define WAIT_ALL()    asm volatile("s_wait_asynccnt 0x0" ::: "memory")
#define SLOT(ch_) shp[(((ch_) - lo) & (DEPTH_ - 1)) * 256 + threadIdx.x]

    // prologue: fill the pipe (hi-lo >= 50 > PRE_, all in-bounds)
    for (int k = 0; k < PRE_; ++k) ISSUE();

    const f4 z = {0.f, 0.f, 0.f, 0.f};
    f4 r0 = z, r1 = z, r2, r3, S;

    // warm-up ring: channels c0-2 .. c0+1 (zeros outside [0,C)); steady waits
    if (c0 >= 2) {
        WAIT_STEADY(); r0 = SLOT(c0 - 2); ISSUE();
        WAIT_STEADY(); r1 = SLOT(c0 - 1); ISSUE();
    }
    WAIT_STEADY(); r2 = SLOT(c0);     ISSUE();
    WAIT_STEADY(); r3 = SLOT(c0 + 1); ISSUE();

    S = r0 * r0 + r1 * r1 + r2 * r2 + r3 * r3;

    // one output channel: window sum, pow via exp2/log2, NT store, slide ring
#define STEP(c_, r4_) do {                                                         \
        f4 Sw_ = S + (r4_) * (r4_);                                                \
        f4 t_  = Sw_ * LRN_K_ + 1.0f;   /* t >= 1, safe for log2 */                \
        f4 e_;                                                                     \
        e_.x = __builtin_amdgcn_exp2f(-BETA_ * __builtin_amdgcn_logf(t_.x));       \
        e_.y = __builtin_amdgcn_exp2f(-BETA_ * __builtin_amdgcn_logf(t_.y));       \
        e_.z = __builtin_amdgcn_exp2f(-BETA_ * __builtin_amdgcn_logf(t_.z));       \
        e_.w = __builtin_amdgcn_exp2f(-BETA_ * __builtin_amdgcn_logf(t_.w));       \
        f4 o_ = r2 * e_;                                                           \
        __builtin_nontemporal_store(o_, ov + base4 + (long)(c_) * HW4_);           \
        S  = Sw_ - r0 * r0;                                                        \
        r0 = r1; r1 = r2; r2 = r3; r3 = (r4_);                                     \
    } while (0)

    // steady state: wait-one, ds-read, issue; invariant nextIssue == (c+2) + PRE_
#pragma unroll 2
    for (int c = c0; c < cs - 1; ++c) {
        WAIT_STEADY();
        f4 r4 = SLOT(c + 2);
        ISSUE();
        STEP(c, r4);
    }
    // peeled last steady iteration: nextIssue == hi, nothing left to issue
    {
        WAIT_STEADY();
        f4 r4 = SLOT(cs + 1);
        STEP(cs - 1, r4);
    }
    // drain: all remaining channels resident in LDS
    WAIT_ALL();
    for (int c = cs; c < c1; ++c) {
        f4 r4 = (c + 2 < C_) ? SLOT(c + 2) : z;
        STEP(c, r4);
    }

#undef ISSUE
#undef WAIT_STEADY
#undef WAIT_ALL
#undef SLOT
#undef STEP
}

extern "C" void kernel_launch(void* const* d_in, const int* in_sizes, int n_in,
                              void* d_out, int out_size, void* d_ws, size_t ws_size,
                              hipStream_t stream) {
    const float* x = (const float*)d_in[0];
    float* out     = (float*)d_out;
    // 64*784 float4 columns / 256 threads = 196 blocks; 4 channel chunks
    dim3 grid((N_ * HW4_) / 256, NCHUNK_, 1);
    lrn_kernel<<<grid, dim3(256, 1, 1), 0, stream>>>(x, out);
}